// GNN_18124761989696
// MI455X (gfx1250) — compile-verified
//
#include <hip/hip_runtime.h>

#define NN 100000
#define IN_CH 128
#define HID 64

typedef __attribute__((ext_vector_type(2))) float v2f;
typedef __attribute__((ext_vector_type(8))) float v8f;
typedef __attribute__((ext_vector_type(4))) unsigned int v4u;
typedef __attribute__((ext_vector_type(8))) int v8i;
typedef __attribute__((ext_vector_type(4))) int v4i;

#if defined(__gfx1250__) && __has_builtin(__builtin_amdgcn_tensor_load_to_lds)
#define USE_TDM 1
#else
#define USE_TDM 0
#endif

// ---------------- utility fills ----------------
__global__ void fill_kernel(float* __restrict__ p, long n, float v) {
    long i = (long)blockIdx.x * blockDim.x + threadIdx.x;
    long stride = (long)gridDim.x * blockDim.x;
    for (; i < n; i += stride) p[i] = v;
}

// ---------------- degree + rsqrt ----------------
__global__ void deg_kernel(const long long* __restrict__ dst,
                           float* __restrict__ deg, int E) {
    int e = blockIdx.x * blockDim.x + threadIdx.x;
    if (e < E) atomicAdd(deg + (int)dst[e], 1.0f);
}

__global__ void dinv_kernel(float* __restrict__ deg, int n) {
    int i = blockIdx.x * blockDim.x + threadIdx.x;
    if (i < n) {
        float d = deg[i];
        deg[i] = (d > 0.0f) ? rsqrtf(d) : 0.0f;
    }
}

// ---------------- GEMM1: h[N,64] = x[N,128] @ W1[128,64] via f32 WMMA ----------------
// Block = 128 threads = 4 waves. Wave w owns rows [blk*64 + w*16, +16), all 64 cols.
// W1 (32 KB) staged into LDS via the Tensor Data Mover (one DMA, wave 0 only),
// then K-loop in steps of 4 using V_WMMA_F32_16X16X4_F32.
__global__ __launch_bounds__(128) void gemm1_wmma(const float* __restrict__ x,
                                                  const float* __restrict__ W1,
                                                  float* __restrict__ h, int N) {
    __shared__ float Ws[IN_CH * HID]; // 32 KB at LDS offset 0

#if USE_TDM
    if (threadIdx.x < 32) { // one wave issues the DMA (TDM ignores EXEC; wave-branch guards it)
        const unsigned long long ga = (unsigned long long)W1;
        const unsigned NE = IN_CH * HID; // 8192 elements of 4 B
        // D# group 0: count=1, lds_addr=0, global_addr, type=2 ("image")
        v4u g0 = { 1u,
                   0u,
                   (unsigned)(ga & 0xFFFFFFFFu),
                   (unsigned)((ga >> 32) & 0x01FFFFFFu) | (2u << 30) };
        // D# group 1: data_size=4B(code 2); tensor_dim0=tile_dim0=stride0=NE; dim1=tile1=1
        v8i g1 = { (int)(2u << 16),                 // wg_mask=0 | data_size=2<<16
                   (int)((NE & 0xFFFFu) << 16),     // bar_addr=0 | tensor_dim0[15:0]
                   (int)(((NE >> 16) & 0xFFFFu) | (1u << 16)), // tensor_dim0[31:16] | tensor_dim1[15:0]=1
                   (int)((NE & 0xFFFFu) << 16),     // tensor_dim1[31:16]=0 | tile_dim0
                   1,                               // tile_dim1=1 | tile_dim2=0
                   (int)NE,                         // tensor_dim0_stride[31:0]
                   0,                               // stride0[47:32] | stride1[15:0]
                   0 };                             // stride1[47:16]
        v4i gz4 = { 0, 0, 0, 0 };
#if __clang_major__ >= 23
        v8i gz8 = { 0, 0, 0, 0, 0, 0, 0, 0 };
        __builtin_amdgcn_tensor_load_to_lds(g0, g1, gz4, gz4, gz8, 0);
#else
        __builtin_amdgcn_tensor_load_to_lds(g0, g1, gz4, gz4, 0);
#endif
        __builtin_amdgcn_s_wait_tensorcnt((short)0);
    }
    // Compiler fence: the TDM's LDS write is invisible to alias analysis. Make Ws
    // escape into an asm with a memory clobber so its reads aren't folded to undef.
    {
        float* wsp = &Ws[0];
        asm volatile("" : "+v"(wsp) : : "memory");
    }
    __syncthreads();
#else
    for (int i = threadIdx.x; i < IN_CH * HID; i += 128) Ws[i] = W1[i];
    __syncthreads();
#endif

    const int lane  = threadIdx.x & 31;
    const int wave  = threadIdx.x >> 5;
    const int rowBase = blockIdx.x * 64 + wave * 16;

    // A layout (16x4 f32): lanes 0-15 -> M=lane, K={0,1}; lanes 16-31 -> M=lane-16, K={2,3}
    const int mrow  = lane & 15;
    const int khalf = (lane >> 4) * 2; // 0 or 2
    int arow = rowBase + mrow;
    if (arow >= N) arow = N - 1;                 // clamp (stores are guarded)
    const float* __restrict__ xrow = x + (size_t)arow * IN_CH;

    // B layout (4x16 f32): lanes 0-15 -> N=lane, K={0,1}; lanes 16-31 -> N=lane-16, K={2,3}
    const int col = lane & 15;

    v8f acc0 = {}, acc1 = {}, acc2 = {}, acc3 = {};

#pragma unroll
    for (int k = 0; k < IN_CH; k += 4) {
        v2f a = *(const v2f*)(xrow + k + khalf);
        const float* w0 = &Ws[(k + khalf) * HID + col];
        const float* w1 = w0 + HID;
        v2f b0, b1, b2, b3;
        b0.x = w0[0];  b0.y = w1[0];
        b1.x = w0[16]; b1.y = w1[16];
        b2.x = w0[32]; b2.y = w1[32];
        b3.x = w0[48]; b3.y = w1[48];
        acc0 = __builtin_amdgcn_wmma_f32_16x16x4_f32(false, a, false, b0, (short)0, acc0, false, false);
        acc1 = __builtin_amdgcn_wmma_f32_16x16x4_f32(false, a, false, b1, (short)0, acc1, false, false);
        acc2 = __builtin_amdgcn_wmma_f32_16x16x4_f32(false, a, false, b2, (short)0, acc2, false, false);
        acc3 = __builtin_amdgcn_wmma_f32_16x16x4_f32(false, a, false, b3, (short)0, acc3, false, false);
    }

    // C/D layout (16x16 f32): VGPR r -> M = r + (lane>=16 ? 8 : 0), N = lane&15
    const int rhalf = (lane >> 4) * 8;
#pragma unroll
    for (int r = 0; r < 8; ++r) {
        int row = rowBase + rhalf + r;
        if (row < N) {
            float* hr = h + (size_t)row * HID + col;
            hr[0]  = acc0[r];
            hr[16] = acc1[r];
            hr[32] = acc2[r];
            hr[48] = acc3[r];
        }
    }
}

// ---------------- Edge scatter, layer 1 (64 channels, 16 threads/edge) ----------------
__global__ void scatter1_kernel(const long long* __restrict__ src,
                                const long long* __restrict__ dst,
                                const float* __restrict__ dinv,
                                const float* __restrict__ h,
                                float* __restrict__ agg, int E) {
    long t = (long)blockIdx.x * blockDim.x + threadIdx.x;
    int e  = (int)(t >> 4);
    int c4 = ((int)t & 15) * 4;
    if (e >= E) return;
    int s = (int)src[e], d = (int)dst[e];
    float w = dinv[s] * dinv[d];
    const float4 hv = *(const float4*)(h + (size_t)s * HID + c4);
    float* o = agg + (size_t)d * HID + c4;
    atomicAdd(o + 0, hv.x * w);
    atomicAdd(o + 1, hv.y * w);
    atomicAdd(o + 2, hv.z * w);
    atomicAdd(o + 3, hv.w * w);
}

// ---------------- Finalize layer 1: h1 = relu(agg + h*dinv^2 + b1) (in place on agg) ----------------
__global__ void finalize1_kernel(float* __restrict__ agg,
                                 const float* __restrict__ h,
                                 const float* __restrict__ dinv,
                                 const float* __restrict__ b1, int N) {
    long t = (long)blockIdx.x * blockDim.x + threadIdx.x;
    int i  = (int)(t >> 4);
    int c4 = ((int)t & 15) * 4;
    if (i >= N) return;
    float di = dinv[i];
    float sl = di * di; // self-loop weight
    float4 a = *(const float4*)(agg + (size_t)i * HID + c4);
    float4 hv = *(const float4*)(h + (size_t)i * HID + c4);
    float r0 = a.x + hv.x * sl + b1[c4 + 0];
    float r1 = a.y + hv.y * sl + b1[c4 + 1];
    float r2 = a.z + hv.z * sl + b1[c4 + 2];
    float r3 = a.w + hv.w * sl + b1[c4 + 3];
    float4 o;
    o.x = fmaxf(r0, 0.0f);
    o.y = fmaxf(r1, 0.0f);
    o.z = fmaxf(r2, 0.0f);
    o.w = fmaxf(r3, 0.0f);
    *(float4*)(agg + (size_t)i * HID + c4) = o;
}

// ---------------- GEMM2: g[N,2] = h1[N,64] @ W2[64,2] ----------------
__global__ __launch_bounds__(256) void gemm2_kernel(const float* __restrict__ h1,
                                                    const float* __restrict__ W2,
                                                    float* __restrict__ g, int N) {
    __shared__ float Ws[HID * 2];
    if (threadIdx.x < HID * 2) Ws[threadIdx.x] = W2[threadIdx.x];
    __syncthreads();
    int i = blockIdx.x * blockDim.x + threadIdx.x;
    if (i >= N) return;
    const float* hr = h1 + (size_t)i * HID;
    float s0 = 0.0f, s1 = 0.0f;
#pragma unroll
    for (int c = 0; c < HID; c += 4) {
        float4 v = *(const float4*)(hr + c);
        s0 += v.x * Ws[(c + 0) * 2] + v.y * Ws[(c + 1) * 2] +
              v.z * Ws[(c + 2) * 2] + v.w * Ws[(c + 3) * 2];
        s1 += v.x * Ws[(c + 0) * 2 + 1] + v.y * Ws[(c + 1) * 2 + 1] +
              v.z * Ws[(c + 2) * 2 + 1] + v.w * Ws[(c + 3) * 2 + 1];
    }
    g[(size_t)i * 2 + 0] = s0;
    g[(size_t)i * 2 + 1] = s1;
}

// ---------------- Edge scatter, layer 2 (2 channels, 1 thread/edge) ----------------
__global__ void scatter2_kernel(const long long* __restrict__ src,
                                const long long* __restrict__ dst,
                                const float* __restrict__ dinv,
                                const float* __restrict__ g,
                                float* __restrict__ out, int E) {
    int e = blockIdx.x * blockDim.x + threadIdx.x;
    if (e >= E) return;
    int s = (int)src[e], d = (int)dst[e];
    float w = dinv[s] * dinv[d];
    float g0 = g[(size_t)s * 2 + 0];
    float g1 = g[(size_t)s * 2 + 1];
    atomicAdd(out + (size_t)d * 2 + 0, g0 * w);
    atomicAdd(out + (size_t)d * 2 + 1, g1 * w);
}

// ---------------- Finalize layer 2: out += g*dinv^2 + b2 ----------------
__global__ void finalize2_kernel(float* __restrict__ out,
                                 const float* __restrict__ g,
                                 const float* __restrict__ dinv,
                                 const float* __restrict__ b2, int N) {
    int i = blockIdx.x * blockDim.x + threadIdx.x;
    if (i >= N) return;
    float di = dinv[i];
    float sl = di * di;
    out[(size_t)i * 2 + 0] += g[(size_t)i * 2 + 0] * sl + b2[0];
    out[(size_t)i * 2 + 1] += g[(size_t)i * 2 + 1] * sl + b2[1];
}

extern "C" void kernel_launch(void* const* d_in, const int* in_sizes, int n_in,
                              void* d_out, int out_size, void* d_ws, size_t ws_size,
                              hipStream_t stream) {
    const float*     x   = (const float*)d_in[0];
    const long long* ei  = (const long long*)d_in[1]; // int64 [2,E]
    const float*     W1  = (const float*)d_in[2];
    const float*     b1  = (const float*)d_in[3];
    const float*     W2  = (const float*)d_in[4];
    const float*     b2  = (const float*)d_in[5];
    float*           out = (float*)d_out;

    const int N = NN;
    const int E = in_sizes[1] / 2;
    const long long* src = ei;
    const long long* dst = ei + E;

    // workspace layout (floats): deg/dinv | h | agg(->h1) | g
    float* ws   = (float*)d_ws;
    float* deg  = ws;                       // N
    float* h    = ws  + 100000;             // N*64
    float* agg  = h   + 6400000;            // N*64
    float* g    = agg + 6400000;            // N*2

    const int T = 256;

    // init: deg=1 (self-loop), agg=0, out=0
    fill_kernel<<<(N + T - 1) / T, T, 0, stream>>>(deg, (long)N, 1.0f);
    fill_kernel<<<8192, T, 0, stream>>>(agg, (long)N * HID, 0.0f);
    fill_kernel<<<(N * 2 + T - 1) / T, T, 0, stream>>>(out, (long)N * 2, 0.0f);

    // degree + dinv
    deg_kernel<<<(E + T - 1) / T, T, 0, stream>>>(dst, deg, E);
    dinv_kernel<<<(N + T - 1) / T, T, 0, stream>>>(deg, N);

    // layer 1
    gemm1_wmma<<<(N + 63) / 64, 128, 0, stream>>>(x, W1, h, N);
    {
        long threads = (long)E * 16;
        scatter1_kernel<<<(unsigned)((threads + T - 1) / T), T, 0, stream>>>(src, dst, deg, h, agg, E);
    }
    {
        long threads = (long)N * 16;
        finalize1_kernel<<<(unsigned)((threads + T - 1) / T), T, 0, stream>>>(agg, h, deg, b1, N);
    }

    // layer 2
    gemm2_kernel<<<(N + T - 1) / T, T, 0, stream>>>(agg, W2, g, N);
    scatter2_kernel<<<(E + T - 1) / T, T, 0, stream>>>(src, dst, deg, g, out, E);
    finalize2_kernel<<<(N + T - 1) / T, T, 0, stream>>>(out, g, deg, b2, N);
}